// SudokuSolver_45440753992522
// MI455X (gfx1250) — compile-verified
//
#include <hip/hip_runtime.h>
#include <hip/hip_bf16.h>
#include <math.h>

typedef float v2f __attribute__((ext_vector_type(2)));
typedef float v8f __attribute__((ext_vector_type(8)));

#define SN    9          // digits
#define SNN   81         // cells
#define SH    100        // hidden
#define SK    27         // 3*n features
#define M_PAD 96         // 81 -> 6 tiles of 16
#define K1PAD 28         // 27 -> 7 steps of 4
#define H_PAD 112        // 100 -> 7 tiles of 16
#define F_S   28         // f row stride (floats): r*28 mod 64 distinct for 16 rows
#define H_S   113        // h row stride (odd -> conflict-free column reads)
#define LOG_S 16

// ---- dynamic LDS layout (float offsets) ----
#define OFF_F    0
#define SZ_F     (M_PAD * F_S)            // 2688
#define OFF_W1T  (OFF_F + SZ_F)
#define SZ_W1T   (K1PAD * H_PAD)          // 3136
#define OFF_W2T  (OFF_W1T + SZ_W1T)
#define SZ_W2T   (H_PAD * LOG_S)          // 1792
#define OFF_H    (OFF_W2T + SZ_W2T)
#define SZ_H     (M_PAD * H_S)            // 10848
#define OFF_LOG  (OFF_H + SZ_H)
#define SZ_LOG   (M_PAD * LOG_S)          // 1536
#define OFF_X    (OFF_LOG + SZ_LOG)
#define SZ_X     (SNN * SN)               // 729
#define OFF_XP   (OFF_X + SZ_X)
#define OFF_GS   (OFF_XP + SZ_X)          // 3*9*9 = 243 group sums
#define OFF_SC   (OFF_GS + 243)           // 96 scores
#define OFF_POS  (OFF_SC + 96)            // 96 best-digit (stored as float)
#define SMEM_FLOATS (OFF_POS + 96)        // = 21893 floats = 87572 B

__global__ __launch_bounds__(256)
void SudokuSolver_kernel(const float* __restrict__ gx,
                         const float* __restrict__ gW1,
                         const float* __restrict__ gW2,
                         float* __restrict__ out_xpred,
                         float* __restrict__ out_x)
{
    extern __shared__ float smem[];
    float* f   = smem + OFF_F;
    float* W1T = smem + OFF_W1T;   // [K1PAD][H_PAD] = W1 transposed, zero-padded
    float* W2T = smem + OFF_W2T;   // [H_PAD][LOG_S] = W2 transposed, zero-padded
    float* h   = smem + OFF_H;     // [M_PAD][H_S] relu activations
    float* lg  = smem + OFF_LOG;   // [M_PAD][LOG_S] logits
    float* x   = smem + OFF_X;     // [81][9] board state
    float* xp  = smem + OFF_XP;    // [81][9] x_pred carry
    float* gs  = smem + OFF_GS;    // [3][9][9] group sums
    float* sc  = smem + OFF_SC;    // per-cell score
    float* pf  = smem + OFF_POS;   // per-cell best digit (float-encoded)

    const int b    = blockIdx.x;
    const int tid  = threadIdx.x;
    const int lane = tid & 31;
    const int wave = tid >> 5;
    const int half = lane >> 4;    // selects K+2 / rows+8 per WMMA f32 layout
    const int lr   = lane & 15;

    // ---------------- one-time init ----------------
    for (int i = tid; i < SZ_F; i += 256) f[i] = 0.0f;   // pads must stay zero
    for (int i = tid; i < SZ_W1T; i += 256) {
        int k = i / H_PAD, c = i % H_PAD;
        W1T[i] = (k < SK && c < SH) ? gW1[c * SK + k] : 0.0f;
    }
    for (int i = tid; i < SZ_W2T; i += 256) {
        int k = i / LOG_S, v = i % LOG_S;
        W2T[i] = (k < SH && v < SN) ? gW2[v * SH + k] : 0.0f;
    }
    const float* xin = gx + (size_t)b * (SNN * SN);
    for (int i = tid; i < SNN * SN; i += 256) {
        float val = xin[i];
        x[i]  = val;
        xp[i] = val;   // carry starts as (x, x)
    }
    __syncthreads();

    for (int iter = 0; iter < SNN; ++iter) {
        // --- phase 1: 243 group sums (row/col/box x digit) ---
        if (tid < 243) {
            int t = tid / 81;
            int g = (tid % 81) / 9;
            int v = tid % 9;
            float s = 0.0f;
            #pragma unroll
            for (int j = 0; j < 9; ++j) {
                int cell;
                if (t == 0)      cell = g * 9 + j;
                else if (t == 1) cell = j * 9 + g;
                else             cell = ((g / 3) * 3 + j / 3) * 9 + ((g % 3) * 3 + j % 3);
                s += x[cell * SN + v];
            }
            gs[tid] = s;
        }
        __syncthreads();

        // --- phase 2: scatter group sums into f[81][27] ---
        for (int i = tid; i < SNN * SK; i += 256) {
            int cell = i / SK;
            int c    = i % SK;
            int t = c / 9, v = c % 9;
            int g = (t == 0) ? (cell / 9)
                  : (t == 1) ? (cell % 9)
                  : ((cell / 27) * 3 + (cell % 9) / 3);
            f[cell * F_S + c] = gs[t * 81 + g * 9 + v];
        }
        __syncthreads();

        // --- phase 3: GEMM1  h = relu(f @ W1T)  (96x112, K=28) ---
        // Prefetch all 7 K-step fragments, then issue WMMAs back-to-back
        // so the whole chain sits behind a single ds wait.
        for (int t = wave; t < 42; t += 8) {
            int mt = t / 7, nt = t % 7;
            int arow = mt * 16 + lr;
            int bcol = nt * 16 + lr;
            v2f afr[7], bfr[7];
            #pragma unroll
            for (int kk = 0; kk < 7; ++kk) {
                int kA = kk * 4 + half * 2;
                afr[kk].x = f[arow * F_S + kA];
                afr[kk].y = f[arow * F_S + kA + 1];
                bfr[kk].x = W1T[kA * H_PAD + bcol];
                bfr[kk].y = W1T[(kA + 1) * H_PAD + bcol];
            }
            v8f acc = {0.f,0.f,0.f,0.f,0.f,0.f,0.f,0.f};
            #pragma unroll
            for (int kk = 0; kk < 7; ++kk)
                acc = __builtin_amdgcn_wmma_f32_16x16x4_f32(
                        false, afr[kk], false, bfr[kk], (short)0, acc, false, false);
            #pragma unroll
            for (int g = 0; g < 8; ++g) {
                int row = mt * 16 + g + half * 8;
                float val = acc[g];
                h[row * H_S + nt * 16 + lr] = val > 0.0f ? val : 0.0f;
            }
        }
        __syncthreads();

        // --- phase 4: GEMM2  logits = h @ W2T  (96x16, K=112) ---
        // 28 K-steps processed in 4 chunks of 7 (bounds frag registers).
        for (int t = wave; t < 6; t += 8) {
            v8f acc = {0.f,0.f,0.f,0.f,0.f,0.f,0.f,0.f};
            int arow = t * 16 + lr;
            #pragma unroll
            for (int c = 0; c < 4; ++c) {
                v2f afr[7], bfr[7];
                #pragma unroll
                for (int kk = 0; kk < 7; ++kk) {
                    int kA = (c * 7 + kk) * 4 + half * 2;
                    afr[kk].x = h[arow * H_S + kA];
                    afr[kk].y = h[arow * H_S + kA + 1];
                    bfr[kk].x = W2T[kA * LOG_S + lr];
                    bfr[kk].y = W2T[(kA + 1) * LOG_S + lr];
                }
                #pragma unroll
                for (int kk = 0; kk < 7; ++kk)
                    acc = __builtin_amdgcn_wmma_f32_16x16x4_f32(
                            false, afr[kk], false, bfr[kk], (short)0, acc, false, false);
            }
            #pragma unroll
            for (int g = 0; g < 8; ++g) {
                int row = t * 16 + g + half * 8;
                lg[row * LOG_S + lr] = acc[g];
            }
        }
        __syncthreads();

        // --- phase 5: per-cell softmax, empty mask, x_pred, score ---
        if (tid < SNN) {
            int i = tid;
            float sx = 0.0f;
            #pragma unroll
            for (int v = 0; v < 9; ++v) sx += x[i * SN + v];
            bool empty = (sx == 0.0f);

            float m = lg[i * LOG_S];
            #pragma unroll
            for (int v = 1; v < 9; ++v) m = fmaxf(m, lg[i * LOG_S + v]);
            float e[9]; float se = 0.0f;
            #pragma unroll
            for (int v = 0; v < 9; ++v) { e[v] = expf(lg[i * LOG_S + v] - m); se += e[v]; }
            float inv = 1.0f / se;

            float best = -1.0f; int bv = 0;
            #pragma unroll
            for (int v = 0; v < 9; ++v) {
                float sv = e[v] * inv;
                if (empty) xp[i * SN + v] = sv;          // x_pred = where(empty, s_, x_pred)
                if (sv > best) { best = sv; bv = v; }    // strict > keeps first (argmax tie rule)
            }
            sc[i] = empty ? best : 0.0f;                 // s = where(empty, s_, 0)
            pf[i] = (float)(empty ? bv : 0);
        }
        __syncthreads();

        // --- phase 6: wave32-parallel argmax over cells, commit one digit ---
        // Lane-local ascending scan (strict >) + shuffle merge with
        // lowest-index tie-break preserves jnp.argmax first-occurrence rule.
        if (wave == 0) {
            float best = -1.0f; int bi = 0;
            for (int i = lane; i < SNN; i += 32) {
                float s = sc[i];
                if (s > best) { best = s; bi = i; }
            }
            #pragma unroll
            for (int off = 16; off > 0; off >>= 1) {
                float ob = __shfl_down(best, off, 32);
                int   oi = __shfl_down(bi,   off, 32);
                if (ob > best || (ob == best && oi < bi)) { best = ob; bi = oi; }
            }
            // softmax max >= 1/9 > 0, so best>0 <=> has_empty
            if (lane == 0 && best > 0.0f) x[bi * SN + (int)pf[bi]] = 1.0f;
        }
        __syncthreads();
    }

    // ---------------- write outputs: (x_pred, x) ----------------
    float* oxp = out_xpred + (size_t)b * (SNN * SN);
    float* ox  = out_x     + (size_t)b * (SNN * SN);
    for (int i = tid; i < SNN * SN; i += 256) {
        oxp[i] = xp[i];
        ox[i]  = x[i];
    }
}

extern "C" void kernel_launch(void* const* d_in, const int* in_sizes, int n_in,
                              void* d_out, int out_size, void* d_ws, size_t ws_size,
                              hipStream_t stream) {
    (void)n_in; (void)d_ws; (void)ws_size; (void)out_size;
    const float* x  = (const float*)d_in[0];   // (B,81,9)
    // d_in[1] = constraint_mask (81,3,81): structural, hardcoded in-kernel
    const float* W1 = (const float*)d_in[2];   // (100,27)
    const float* W2 = (const float*)d_in[3];   // (9,100)

    int B = in_sizes[0] / (SNN * SN);
    float* out_xpred = (float*)d_out;
    float* out_x     = out_xpred + (size_t)B * SNN * SN;

    size_t smem_bytes = (size_t)SMEM_FLOATS * sizeof(float);
    // allow >64KB dynamic LDS (CDNA5: 320KB per workgroup)
    hipFuncSetAttribute((const void*)SudokuSolver_kernel,
                        hipFuncAttributeMaxDynamicSharedMemorySize,
                        (int)smem_bytes);

    SudokuSolver_kernel<<<B, 256, smem_bytes, stream>>>(x, W1, W2, out_xpred, out_x);
}